// TwoStageAggregator_22789096472918
// MI455X (gfx1250) — compile-verified
//
#include <hip/hip_runtime.h>

// ---------------------------------------------------------------------------
// Two-stage SetTransformer aggregation for gfx1250 (MI455X).
// All GEMMs on V_WMMA_F32_16X16X4_F32.  Explicit address-space typing so the
// backend emits global_load_b64 for weights and ds_load_b64 (immediate-offset)
// for LDS fragments instead of flat/generic traffic.
// ---------------------------------------------------------------------------

#define LDSTR 130   // LDS row stride (floats): 130 mod 64 = 2 -> conflict-free b64 frag loads
#define SSTR  50    // score row stride (floats)

typedef float v2f __attribute__((ext_vector_type(2)));
typedef float v8f __attribute__((ext_vector_type(8)));

typedef __attribute__((address_space(3))) float       LF;   // LDS float
typedef __attribute__((address_space(1))) const float GF;   // global const float
typedef __attribute__((address_space(1))) float       GWf;  // global float (write)
typedef __attribute__((address_space(1))) const int   GIc;  // global const int
typedef __attribute__((address_space(1))) int         GIw;  // global int (write)
typedef __attribute__((address_space(3))) v2f         LV2;
typedef __attribute__((address_space(1))) const v2f   GV2;

struct MabP {   // host-built (generic pointers)
  const float *in_w, *in_b, *out_w, *out_b, *lin_w, *lin_b;
  const float *ln1_g, *ln1_b, *ln2_g, *ln2_b;
};
struct StP {
  MabP enc0, enc1, dec0, dec1, pma;
  const float *pma_lin_w, *pma_lin_b, *seed;
};
struct MabG {   // device-side, AS(1)-typed
  GF *in_w, *in_b, *out_w, *out_b, *lin_w, *lin_b;
  GF *ln1_g, *ln1_b, *ln2_g, *ln2_b;
};
__device__ __forceinline__ MabG cvt(const MabP& p) {
  MabG g;
  g.in_w  = (GF*)p.in_w;  g.in_b  = (GF*)p.in_b;
  g.out_w = (GF*)p.out_w; g.out_b = (GF*)p.out_b;
  g.lin_w = (GF*)p.lin_w; g.lin_b = (GF*)p.lin_b;
  g.ln1_g = (GF*)p.ln1_g; g.ln1_b = (GF*)p.ln1_b;
  g.ln2_g = (GF*)p.ln2_g; g.ln2_b = (GF*)p.ln2_b;
  return g;
}

__device__ __forceinline__ v2f ld2(const LF* p)  { return *(const LV2*)p; }
__device__ __forceinline__ v2f ld2g(GF* p)       { return *(const GV2*)p; }

__device__ __forceinline__ v8f wmma_f32(v2f a, v2f b, v8f c) {
  // D = A(16x4 f32) * B(4x16 f32) + C(16x16 f32)
  return __builtin_amdgcn_wmma_f32_16x16x4_f32(false, a, false, b, (short)0, c,
                                               false, false);
}

enum { ST_PLAIN = 0, ST_RELU = 1, ST_MASK_RES = 2, ST_ADD_RELU = 3 };

// D = A @ W^T + bias.  A: LDS [RT*16][LDSTR], W: global row-major [N][128], K=128.
// Column tile nt -> buffer Dbase + (nt>>3)*seg, column (nt&7)*16 (fused packed
// outputs such as QKV).  Epilogue fused per `mode`.
template <int RT>
__device__ void gemm_wt(const LF* As, GF* W, GF* bias, int Ntiles,
                        LF* Dbase, int seg, int mode, const LF* aux, int L)
{
  const int lane = threadIdx.x & 31;
  const int wv   = threadIdx.x >> 5;
  const int cit  = lane & 15;           // column-in-tile (and A row selector)
  const int ksel = (lane >> 4) << 1;    // lanes 16-31 carry K+2,K+3
  const int rsh  = (lane >> 4) << 3;    // C/D: lanes 16-31 hold rows M+8
  const LF* abase = As + cit*LDSTR + ksel;
  for (int nt = wv; nt < Ntiles; nt += 8) {
    const int ncol = nt * 16 + cit;
    const float bv = bias[ncol];
    v8f acc[RT];
#pragma unroll
    for (int mt = 0; mt < RT; ++mt) acc[mt] = v8f{bv,bv,bv,bv,bv,bv,bv,bv};
    GF* wrow = W + ncol * 128 + ksel;
#pragma unroll
    for (int k0 = 0; k0 < 128; k0 += 4) {
      const v2f b = ld2g(wrow + k0);
#pragma unroll
      for (int mt = 0; mt < RT; ++mt)
        acc[mt] = wmma_f32(ld2(abase + mt*(16*LDSTR) + k0), b, acc[mt]);
    }
    LF* Dp = Dbase + (nt >> 3) * seg + ((nt & 7)*16 + cit) + rsh*LDSTR;
    const LF* ax = aux + ((nt & 7)*16 + cit) + rsh*LDSTR;
#pragma unroll
    for (int mt = 0; mt < RT; ++mt) {
#pragma unroll
      for (int r = 0; r < 8; ++r) {
        const int row = mt*16 + rsh + r;
        const float v = acc[mt][r];
        float o;
        if (mode == ST_PLAIN)          o = v;
        else if (mode == ST_RELU)      o = fmaxf(v, 0.f);
        else if (mode == ST_MASK_RES)  o = (row < L ? v : 0.f) + ax[(mt*16 + r)*LDSTR];
        else /* ST_ADD_RELU */         o = ax[(mt*16 + r)*LDSTR] + fmaxf(v, 0.f);
        Dp[(mt*16 + r)*LDSTR] = o;
      }
    }
  }
}

// Scores for 4 heads (pass p: heads 4p..4p+3):
// S[hl][i][j] = 0.25 * sum_d q[i][16h+d]*k[j][16h+d]  (pad keys -> -1e9)
template <int RT>
__device__ void attn_scores4(const LF* q, const LF* k, LF* S, int L, int p)
{
  const int R = RT*16;
  const int lane = threadIdx.x & 31;
  const int wv   = threadIdx.x >> 5;
  const int cit  = lane & 15;
  const int ksel = (lane >> 4) << 1;
  const int rsh  = (lane >> 4) << 3;
  for (int t = wv; t < 4*RT*RT; t += 8) {
    const int hl  = t / (RT*RT);
    const int rem = t % (RT*RT);
    const int mt = rem / RT, nt = rem % RT;
    const int h  = 4*p + hl;
    v8f acc = v8f{0,0,0,0,0,0,0,0};
    const LF* qp = q + (mt*16 + cit)*LDSTR + h*16 + ksel;
    const LF* kp = k + (nt*16 + cit)*LDSTR + h*16 + ksel;
#pragma unroll
    for (int k0 = 0; k0 < 16; k0 += 4)
      acc = wmma_f32(ld2(qp + k0), ld2(kp + k0), acc);   // B[d][j] = k[j][d]
    const int j = nt*16 + cit;
    const bool valid = (j < L);
    LF* Sp = S + (hl*R + mt*16 + rsh) * SSTR + j;
#pragma unroll
    for (int r = 0; r < 8; ++r)
      Sp[r*SSTR] = valid ? acc[r]*0.25f : -1e9f;
  }
}

template <int RT>
__device__ void softmax_S4(LF* S)
{
  const int R = RT*16;
  for (int row = threadIdx.x; row < 4*R; row += 256) {
    LF* pr = S + row * SSTR;
    float m = pr[0];
    for (int j = 1; j < R; ++j) m = fmaxf(m, pr[j]);
    float s = 0.f;
    for (int j = 0; j < R; ++j) { const float e = __expf(pr[j] - m); pr[j] = e; s += e; }
    const float inv = 1.f / s;
    for (int j = 0; j < R; ++j) pr[j] *= inv;
  }
}

// O[:,16h:16h+16] = S_hl @ V_h  — 2 waves per head, M-tiles split across them.
template <int RT>
__device__ void attn_o4(const LF* S, const LF* v, LF* O, int p)
{
  const int R = RT*16;
  const int lane = threadIdx.x & 31;
  const int wv   = threadIdx.x >> 5;
  const int hl   = wv & 3;
  const int h    = 4*p + hl;
  const int cit  = lane & 15;
  const int ksel = (lane >> 4) << 1;
  const int rsh  = (lane >> 4) << 3;
  for (int mt = (wv >> 2); mt < RT; mt += 2) {
    v8f acc = v8f{0,0,0,0,0,0,0,0};
    const LF* Sp = S + (hl*R + mt*16 + cit)*SSTR + ksel;
#pragma unroll
    for (int k0 = 0; k0 < R; k0 += 4) {
      v2f a = ld2(Sp + k0);
      v2f b;
      b.x = v[(k0 + ksel    )*LDSTR + h*16 + cit];
      b.y = v[(k0 + ksel + 1)*LDSTR + h*16 + cit];
      acc = wmma_f32(a, b, acc);
    }
#pragma unroll
    for (int r = 0; r < 8; ++r)
      O[(mt*16 + rsh + r)*LDSTR + h*16 + cit] = acc[r];
  }
}

// Per-row LayerNorm over 128 columns; one wave per row, shfl reductions.
__device__ void ln_rows(LF* buf, int R, GF* g, GF* b)
{
  const int lane = threadIdx.x & 31;
  const int wv   = threadIdx.x >> 5;
  for (int row = wv; row < R; row += 8) {
    float v0[4]; float s = 0.f, s2 = 0.f;
#pragma unroll
    for (int i = 0; i < 4; ++i) {
      const float x = buf[row*LDSTR + lane + 32*i];
      v0[i] = x; s += x; s2 += x*x;
    }
#pragma unroll
    for (int off = 16; off > 0; off >>= 1) {
      s  += __shfl_xor(s,  off, 32);
      s2 += __shfl_xor(s2, off, 32);
    }
    const float mu  = s * (1.f/128.f);
    const float var = fmaxf(s2 * (1.f/128.f) - mu*mu, 0.f);
    const float inv = rsqrtf(var + 1e-5f);
#pragma unroll
    for (int i = 0; i < 4; ++i) {
      const int c = lane + 32*i;
      buf[row*LDSTR + c] = (v0[i] - mu) * inv * g[c] + b[c];
    }
  }
}

// LayerNorm of one 128-vector in LDS.
__device__ void ln_vec(LF* v, GF* g, GF* b, LF* red)
{
  if (threadIdx.x < 32) {
    float s = 0.f, s2 = 0.f;
#pragma unroll
    for (int i = 0; i < 4; ++i) { const float x = v[threadIdx.x + 32*i]; s += x; s2 += x*x; }
#pragma unroll
    for (int off = 16; off > 0; off >>= 1) {
      s += __shfl_xor(s, off, 32); s2 += __shfl_xor(s2, off, 32);
    }
    if (threadIdx.x == 0) { red[0] = s*(1.f/128.f); red[1] = s2*(1.f/128.f); }
  }
  __syncthreads();
  const float mu  = red[0];
  const float var = fmaxf(red[1] - mu*mu, 0.f);
  const float inv = rsqrtf(var + 1e-5f);
  if (threadIdx.x < 128)
    v[threadIdx.x] = (v[threadIdx.x] - mu)*inv*g[threadIdx.x] + b[threadIdx.x];
  __syncthreads();
}

// One self-attention MAB in LDS.  Xs updated in place; qkv = 3 scratch tiles.
template <int RT>
__device__ void mab_self(LF* Xs, LF* qkv, LF* S, const MabG p, int L)
{
  const int R = RT*16;
  const int BUF = R*LDSTR;
  gemm_wt<RT>(Xs, p.in_w, p.in_b, 24, qkv, BUF, ST_PLAIN, Xs, L);        // packed Q,K,V
  __syncthreads();
  for (int pass = 0; pass < 2; ++pass) {         // 4 heads per pass; O overwrites Q cols
    attn_scores4<RT>(qkv, qkv + BUF, S, L, pass);
    __syncthreads();
    softmax_S4<RT>(S);
    __syncthreads();
    attn_o4<RT>(S, qkv + 2*BUF, qkv, pass);
    __syncthreads();
  }
  gemm_wt<RT>(qkv, p.out_w, p.out_b, 8, qkv + BUF, 0, ST_MASK_RES, Xs, L); // +x_mask,+res
  __syncthreads();
  ln_rows(qkv + BUF, R, p.ln1_g, p.ln1_b);
  __syncthreads();
  gemm_wt<RT>(qkv + BUF, p.lin_w, p.lin_b, 8, Xs, 0, ST_ADD_RELU, qkv + BUF, R);
  __syncthreads();
  ln_rows(Xs, R, p.ln2_g, p.ln2_b);
  __syncthreads();
}

// Encoders + PMA pooling for one segment (molecule / blend) per workgroup.
template <int RT>
__global__ void __launch_bounds__(256)
st_kernel(const float* xin_, const int* starts_, StP sp, float* pooled_)
{
  constexpr int R = RT*16;
  constexpr int BUF = R*LDSTR;
  __shared__ float smem_[4*R*LDSTR + 4*R*SSTR];
  LF* smem = (LF*)smem_;
  LF* Xs  = smem;            // [R][LDSTR]
  LF* qkv = Xs + BUF;        // 3 x [R][LDSTR]
  LF* S   = qkv + 3*BUF;     // [4][R][SSTR]  (also scalar scratch in PMA)
  GF*  xin    = (GF*)xin_;
  GIc* starts = (GIc*)starts_;
  GWf* pooled = (GWf*)pooled_;

  const int seg   = blockIdx.x;
  const int start = starts[seg];
  int L = starts[seg+1] - start;
  if (L > R) L = R;

  for (int idx = threadIdx.x; idx < R*128; idx += 256) {
    const int r = idx >> 7, c = idx & 127;
    Xs[r*LDSTR + c] = (r < L) ? xin[(start + r)*128 + c] : 0.f;
  }
  __syncthreads();

  mab_self<RT>(Xs, qkv, S, cvt(sp.enc0), L);
  mab_self<RT>(Xs, qkv, S, cvt(sp.enc1), L);

  // ---- PMA:  H = relu(X @ plin^T + b); seed attends over H (key mask) ----
  gemm_wt<RT>(Xs, (GF*)sp.pma_lin_w, (GF*)sp.pma_lin_b, 8, qkv, 0, ST_RELU, Xs, R); // H
  __syncthreads();
  const MabG pm = cvt(sp.pma);
  gemm_wt<RT>(qkv, pm.in_w + 128*128, pm.in_b + 128, 16, qkv + BUF, BUF,
              ST_PLAIN, Xs, R);                                                     // K,V
  __syncthreads();

  LF* qvec = S;          // [128]
  LF* ovec = S + 128;    // [128]
  LF* uvec = S + 256;    // [128]
  LF* zvec = S + 384;    // [128]
  LF* red  = S + 512;    // [2]
  LF* att  = S + 544;    // [8*R]
  const LF* kb = qkv + BUF;
  const LF* vb = qkv + 2*BUF;
  GF* seed = (GF*)sp.seed;
  const int tid = threadIdx.x;

  if (tid < 128) {                                   // q = seed @ Wq^T + bq
    float s = pm.in_b[tid];
    GF* w = pm.in_w + tid*128;
    for (int c = 0; c < 128; ++c) s += w[c] * seed[c];
    qvec[tid] = s;
  }
  __syncthreads();
  for (int idx = tid; idx < 8*R; idx += 256) {       // scores (masked)
    const int h = idx / R, j = idx % R;
    float s;
    if (j < L) {
      s = 0.f;
      const LF* qh = qvec + h*16;
      const LF* kr = kb + j*LDSTR + h*16;
      for (int d = 0; d < 16; ++d) s += qh[d]*kr[d];
      s *= 0.25f;
    } else s = -1e9f;
    att[idx] = s;
  }
  __syncthreads();
  if (tid < 8) {                                     // softmax per head
    LF* pr = att + tid*R;
    float m = pr[0];
    for (int j = 1; j < R; ++j) m = fmaxf(m, pr[j]);
    float s = 0.f;
    for (int j = 0; j < R; ++j) { const float e = __expf(pr[j]-m); pr[j] = e; s += e; }
    const float inv = 1.f/s;
    for (int j = 0; j < R; ++j) pr[j] *= inv;
  }
  __syncthreads();
  if (tid < 128) {                                   // o = a @ v
    const int h = tid >> 4;
    float s = 0.f;
    const LF* a = att + h*R;
    for (int j = 0; j < R; ++j) s += a[j] * vb[j*LDSTR + tid];
    ovec[tid] = s;
  }
  __syncthreads();
  if (tid < 128) {                                   // out proj + residual(seed)
    float s = pm.out_b[tid];
    GF* w = pm.out_w + tid*128;
    for (int c = 0; c < 128; ++c) s += w[c]*ovec[c];
    uvec[tid] = s + seed[tid];
  }
  __syncthreads();
  ln_vec(uvec, pm.ln1_g, pm.ln1_b, red);
  if (tid < 128) {                                   // + relu(lin)
    float s = pm.lin_b[tid];
    GF* w = pm.lin_w + tid*128;
    for (int c = 0; c < 128; ++c) s += w[c]*uvec[c];
    zvec[tid] = uvec[tid] + fmaxf(s, 0.f);
  }
  __syncthreads();
  ln_vec(zvec, pm.ln2_g, pm.ln2_b, red);
  if (tid < 128) pooled[seg*128 + tid] = zvec[tid];
}

// Batched L=1 decoder MABs: softmax over one key == 1  =>  mha(x,x) = (xWv+bv)Wo+bo.
// 64 rows per workgroup; ends with nan_to_num.
__global__ void __launch_bounds__(256)
dec_kernel(const float* in_, StP sp, float* out_)
{
  __shared__ float smem_[3*64*LDSTR];
  LF* B0 = (LF*)smem_;
  LF* B1 = B0 + 64*LDSTR;
  LF* B2 = B1 + 64*LDSTR;
  GF*  in  = (GF*)in_;
  GWf* out = (GWf*)out_;
  const int rowbase = blockIdx.x * 64;
  for (int idx = threadIdx.x; idx < 64*128; idx += 256) {
    const int r = idx >> 7, c = idx & 127;
    B0[r*LDSTR + c] = in[(rowbase + r)*128 + c];
  }
  __syncthreads();
  const MabG decs[2] = { cvt(sp.dec0), cvt(sp.dec1) };
  for (int l = 0; l < 2; ++l) {
    const MabG p = decs[l];
    gemm_wt<4>(B0, p.in_w + 256*128, p.in_b + 256, 8, B1, 0, ST_PLAIN, B0, 64); // V
    __syncthreads();
    gemm_wt<4>(B1, p.out_w, p.out_b, 8, B2, 0, ST_MASK_RES, B0, 64);            // O + x
    __syncthreads();
    ln_rows(B2, 64, p.ln1_g, p.ln1_b);
    __syncthreads();
    gemm_wt<4>(B2, p.lin_w, p.lin_b, 8, B0, 0, ST_ADD_RELU, B2, 64);
    __syncthreads();
    ln_rows(B0, 64, p.ln2_g, p.ln2_b);
    __syncthreads();
  }
  for (int idx = threadIdx.x; idx < 64*128; idx += 256) {
    const int r = idx >> 7, c = idx & 127;
    float v = B0[r*LDSTR + c];
    if (v != v) v = 0.f;
    else if (v >  3.402823466e38f) v =  3.402823466e38f;
    else if (v < -3.402823466e38f) v = -3.402823466e38f;
    out[(rowbase + r)*128 + c] = v;
  }
}

// starts[s] = lower_bound(batch, s);  starts[nseg] = n.
__global__ void starts_kernel(const int* batch_, int n, int nseg, int* starts_)
{
  GIc* batch  = (GIc*)batch_;
  GIw* starts = (GIw*)starts_;
  const int s = blockIdx.x*blockDim.x + threadIdx.x;
  if (s > nseg) return;
  int lo = 0, hi = n;
  while (lo < hi) { const int mid = (lo + hi) >> 1; if (batch[mid] < s) lo = mid + 1; else hi = mid; }
  starts[s] = lo;
}

// ---------------------------------------------------------------------------
// Host side: decode the flattened JAX pytree (dict keys sorted within pytrees;
// top-level entries in setup_inputs() insertion order).
// MAB leaf order: in_b,in_w,lin_b,lin_w,ln1_b,ln1_g,ln2_b,ln2_g,out_b,out_w.
// StP leaf order: dec(10, leading [2]), enc(10, leading [2]),
//                 pma_lin_b, pma_lin_w, pma.mab(10), seed.
// ---------------------------------------------------------------------------
static MabP make_mab(void* const* d, int base, int layer) {
  const size_t l = (layer < 0) ? 0 : (size_t)layer;
  MabP m;
  m.in_b  = (const float*)d[base+0] + l*384;
  m.in_w  = (const float*)d[base+1] + l*384*128;
  m.lin_b = (const float*)d[base+2] + l*128;
  m.lin_w = (const float*)d[base+3] + l*128*128;
  m.ln1_b = (const float*)d[base+4] + l*128;
  m.ln1_g = (const float*)d[base+5] + l*128;
  m.ln2_b = (const float*)d[base+6] + l*128;
  m.ln2_g = (const float*)d[base+7] + l*128;
  m.out_b = (const float*)d[base+8] + l*128;
  m.out_w = (const float*)d[base+9] + l*128*128;
  return m;
}
static StP make_stp(void* const* d, int base) {
  StP s;
  s.dec0 = make_mab(d, base + 0, 0);
  s.dec1 = make_mab(d, base + 0, 1);
  s.enc0 = make_mab(d, base + 10, 0);
  s.enc1 = make_mab(d, base + 10, 1);
  s.pma_lin_b = (const float*)d[base + 20];
  s.pma_lin_w = (const float*)d[base + 21];
  s.pma  = make_mab(d, base + 22, -1);
  s.seed = (const float*)d[base + 32];
  return s;
}

extern "C" void kernel_launch(void* const* d_in, const int* in_sizes, int n_in,
                              void* d_out, int out_size, void* d_ws, size_t ws_size,
                              hipStream_t stream)
{
  (void)n_in; (void)out_size; (void)ws_size;
  const float* x           = (const float*)d_in[0];
  const int*   mol_batch   = (const int*)d_in[1];
  const int*   blend_batch = (const int*)d_in[2];
  const StP molp = make_stp(d_in, 3);
  const StP blp  = make_stp(d_in, 36);

  char* w8 = (char*)d_ws;
  int*   mol_starts   = (int*)(w8);
  int*   blend_starts = (int*)(w8 + (64 << 10));
  float* pooled1      = (float*)(w8 + (128 << 10));
  float* mols         = pooled1 + (size_t)8192*128;
  float* pooled2      = mols    + (size_t)8192*128;

  const int n_nodes = in_sizes[1];   // 262144
  const int n_mols  = in_sizes[2];   // 8192

  starts_kernel<<<(8193 + 255)/256, 256, 0, stream>>>(mol_batch, n_nodes, 8192, mol_starts);
  starts_kernel<<<(513  + 255)/256, 256, 0, stream>>>(blend_batch, n_mols, 512, blend_starts);

  st_kernel<3><<<8192, 256, 0, stream>>>(x, mol_starts, molp, pooled1);
  dec_kernel<<<8192/64, 256, 0, stream>>>(pooled1, molp, mols);

  st_kernel<2><<<512, 256, 0, stream>>>(mols, blend_starts, blp, pooled2);
  dec_kernel<<<512/64, 256, 0, stream>>>(pooled2, blp, (float*)d_out);
}